// LatentResonanceAttentionV18_28638841930301
// MI455X (gfx1250) — compile-verified
//
#include <hip/hip_runtime.h>
#include <hip/hip_bf16.h>
#include <math.h>

// Problem constants (from reference setup_inputs)
#define BB   2
#define LL   512
#define DD   512
#define LAT  256
#define HH   4
#define DHH  128
#define CT   16      // recurrence chunk length
#define EPSN 1e-6f
#define PIF  3.14159265358979f

typedef __attribute__((ext_vector_type(2))) float v2f;
typedef __attribute__((ext_vector_type(8))) float v8f;
typedef __attribute__((ext_vector_type(4))) unsigned int u32x4;
typedef __attribute__((ext_vector_type(8))) int i32x8;
typedef __attribute__((ext_vector_type(4))) int i32x4;

__device__ __forceinline__ float sigmoidf_(float x) { return 1.0f / (1.0f + __expf(-x)); }

#if defined(__has_builtin)
#if __has_builtin(__builtin_amdgcn_tensor_load_to_lds) && __has_builtin(__builtin_amdgcn_s_wait_tensorcnt)
#define USE_TDM 1
#endif
#endif

#ifdef USE_TDM
// Issue one TDM 2D tile load: tile = CT rows x DHH f32 (512B/row), row pitch
// `stride_elems` f32, into LDS byte offset lds_off. D# per ISA ch.8.
// 6-arg builtin form (clang-23 / therock-10.0 headers).
__device__ __forceinline__ void tdm_load_tile(const float* gptr, unsigned lds_off,
                                              unsigned stride_elems) {
  unsigned long long ga = (unsigned long long)(size_t)gptr;
  u32x4 g0;
  g0[0] = 1u;                                            // count=1, user D#
  g0[1] = lds_off;                                       // lds_addr (bytes)
  g0[2] = (unsigned)(ga & 0xFFFFFFFFu);                  // global_addr[31:0]
  g0[3] = (unsigned)((ga >> 32) & 0x1FFFFFFu) | (2u << 30); // addr[56:32] | type=2
  i32x8 g1;
  g1[0] = (int)(2u << 16);                               // data_size=4B
  g1[1] = (int)((unsigned)DHH << 16);                    // tensor_dim0[15:0]
  g1[2] = (int)((unsigned)CT << 16);                     // dim0 hi16=0 | tensor_dim1[15:0]
  g1[3] = (int)((unsigned)DHH << 16);                    // dim1 hi16=0 | tile_dim0
  g1[4] = (int)(unsigned)CT;                             // tile_dim1 | tile_dim2=0
  g1[5] = (int)stride_elems;                             // tensor_dim0_stride[31:0]
  g1[6] = 0;
  g1[7] = 0;
  i32x4 z4; z4[0] = 0; z4[1] = 0; z4[2] = 0; z4[3] = 0;
  i32x8 z8; z8[0] = 0; z8[1] = 0; z8[2] = 0; z8[3] = 0;
            z8[4] = 0; z8[5] = 0; z8[6] = 0; z8[7] = 0;
  __builtin_amdgcn_tensor_load_to_lds(g0, g1, z4, z4, z8, 0);
}
#endif

// ---------------------------------------------------------------------------
// Row-wise RMSNorm: y[row][d] = w[d] * x[row][d] * rsqrt(mean(x^2)+eps)
// ---------------------------------------------------------------------------
__global__ void rmsnorm_rows_kernel(const float* __restrict__ x,
                                    const float* __restrict__ w,
                                    float* __restrict__ y, int Dn) {
  int row = blockIdx.x;
  int tid = threadIdx.x;
  const float* xr = x + (size_t)row * Dn;
  float s = 0.0f;
  for (int d = tid; d < Dn; d += blockDim.x) { float v = xr[d]; s += v * v; }
  __shared__ float red[256];
  red[tid] = s; __syncthreads();
  for (int st = 128; st > 0; st >>= 1) { if (tid < st) red[tid] += red[tid + st]; __syncthreads(); }
  float rms = rsqrtf(red[0] / (float)Dn + EPSN);
  for (int d = tid; d < Dn; d += blockDim.x)
    y[(size_t)row * Dn + d] = w[d] * xr[d] * rms;
}

// ---------------------------------------------------------------------------
// f32 WMMA GEMM: C = A(MxK) @ B(KxN), row-major, M,N mult of 16, K mult of 4.
// 8 waves/block, one 16x16 C tile per wave, V_WMMA_F32_16X16X4_F32.
// mode: 0=none, 1=silu, 2=multiply by aux[row*N+col]
// ---------------------------------------------------------------------------
__global__ void gemm_wmma_f32_kernel(const float* __restrict__ A,
                                     const float* __restrict__ B,
                                     float* __restrict__ C,
                                     const float* __restrict__ aux,
                                     int M, int N, int K, int mode) {
  int lane = threadIdx.x & 31;
  int wave = threadIdx.x >> 5;
  int tilesN = N >> 4;
  int tile = blockIdx.x * 8 + wave;
  int tm = tile / tilesN;
  int tn = tile - tm * tilesN;
  int half = lane >> 4;
  int idx  = lane & 15;

  const float* Arow = A + (size_t)(tm * 16 + idx) * K;
  const float* Bcol = B + (tn * 16 + idx);

  v8f acc = {};
  for (int k0 = 0; k0 < K; k0 += 4) {
    int ka = k0 + half * 2;
    v2f a; a.x = Arow[ka];                 a.y = Arow[ka + 1];
    v2f b; b.x = Bcol[(size_t)ka * N];     b.y = Bcol[(size_t)(ka + 1) * N];
    acc = __builtin_amdgcn_wmma_f32_16x16x4_f32(false, a, false, b, (short)0, acc, false, false);
  }

#pragma unroll
  for (int r = 0; r < 8; ++r) {
    int row = tm * 16 + r + half * 8;
    int col = tn * 16 + idx;
    float v = acc[r];
    if (mode == 1)      v = v / (1.0f + __expf(-v));
    else if (mode == 2) v *= aux[(size_t)row * N + col];
    C[(size_t)row * N + col] = v;
  }
}

// ---------------------------------------------------------------------------
// q/k: per-(b,l,h) RMSNorm(dh) + RoPE(pos=l) + elu(x)+1
// grid.x = B*L*H, grid.y = 0 (q) / 1 (k), block = 128
// ---------------------------------------------------------------------------
__global__ void qk_rope_kernel(const float* __restrict__ qkv,
                               const float* __restrict__ qn_w,
                               const float* __restrict__ kn_w,
                               float* __restrict__ q_f,
                               float* __restrict__ k_f) {
  int row = blockIdx.x;            // (b*L + l)*H + h
  int h   = row & (HH - 1);
  int bl  = row >> 2;              // b*L + l
  int l   = bl & (LL - 1);
  int tid = threadIdx.x;           // 0..127 == d
  bool isK = (blockIdx.y == 1);
  const float* w   = isK ? kn_w : qn_w;
  const float* src = qkv + (size_t)bl * (3 * DD) + (isK ? DD : 0) + h * DHH;

  __shared__ float red[128];
  __shared__ float xn[128];
  float x = src[tid];
  red[tid] = x * x; __syncthreads();
  for (int st = 64; st > 0; st >>= 1) { if (tid < st) red[tid] += red[tid + st]; __syncthreads(); }
  float rms = rsqrtf(red[0] / (float)DHH + EPSN);
  xn[tid] = w[tid] * x * rms;
  __syncthreads();

  int i = tid & 63;
  float fr  = __expf(-(2.0f * (float)i / (float)DHH) * logf(10000.0f));
  float ang = (float)l * fr;
  float c = cosf(ang), s = sinf(ang);
  float r;
  if (tid < 64) r = xn[tid] * c - xn[tid + 64] * s;
  else          r = xn[tid] * c + xn[tid - 64] * s;
  float out = (r > 0.0f) ? (r + 1.0f) : __expf(r);     // elu(r)+1
  float* dst = (isK ? k_f : q_f) + (size_t)row * DHH;
  dst[tid] = out;
}

// ---------------------------------------------------------------------------
// Resonance gate from params (B,L,H,4)
// ---------------------------------------------------------------------------
__global__ void gate_kernel(const float* __restrict__ params,
                            const float* __restrict__ temperature,
                            float* __restrict__ gate) {
  int i = blockIdx.x * blockDim.x + threadIdx.x;
  if (i >= BB * LL * HH) return;
  const float* p = params + (size_t)i * 4;
  float sa = sigmoidf_(p[0]);
  float sp = sigmoidf_(p[1]) * PIF;
  float ca = sigmoidf_(p[2]);
  float cp = sigmoidf_(p[3]) * PIF;
  float base = sigmoidf_(sa * ca * cosf(sp - cp) * temperature[0]);
  float g = base * 1.2f - 0.1f;
  gate[i] = fminf(fmaxf(g, 0.05f), 0.95f);
}

// ---------------------------------------------------------------------------
// Chunked linear-attention recurrence, one workgroup per (b,h), 8 waves.
// LDS: C(128x128), Q/K/V chunk (16x128 each), A/W (16x16), Z(128), coefs.
// Per chunk: A=QK^T (WMMA); out = df*(Q@C + W@V)/den (WMMA); C += K'^T@V (WMMA).
// Q/K/V staged via TDM tensor_load_to_lds when available.
// ---------------------------------------------------------------------------
#define LDS_C_FLOATS   (DHH * DHH)                 // 16384
#define LDS_QB_OFF_B   (LDS_C_FLOATS * 4)          // 65536
#define LDS_KB_OFF_B   (LDS_QB_OFF_B + CT * DHH * 4)  // 73728
#define LDS_VB_OFF_B   (LDS_KB_OFF_B + CT * DHH * 4)  // 81920
#define REC_SMEM_FLOATS (LDS_C_FLOATS + 3 * CT * DHH + 256 + 256 + DHH + 4 * CT)

__global__ void recurrence_chunked_kernel(const float* __restrict__ q_f,
                                          const float* __restrict__ k_f,
                                          const float* __restrict__ qkv,   // v at +2D
                                          const float* __restrict__ gate,
                                          const float* __restrict__ head_decay,
                                          float* __restrict__ attn_raw) {
  extern __shared__ float smem[];
  float* Cbuf = smem;                   // 128*128
  float* Qb   = Cbuf + LDS_C_FLOATS;    // 16*128
  float* Kb   = Qb + CT * DHH;          // 16*128
  float* Vb   = Kb + CT * DHH;          // 16*128
  float* Ab   = Vb + CT * DHH;          // 16*16
  float* Wb   = Ab + 256;               // 16*16
  float* Zb   = Wb + 256;               // 128
  float* cKV  = Zb + DHH;               // 16
  float* cZ   = cKV + CT;               // 16
  float* dfb  = cZ + CT;                // 16
  float* denb = dfb + CT;               // 16

  const int b    = blockIdx.x >> 2;
  const int h    = blockIdx.x & 3;
  const int tid  = threadIdx.x;
  const int lane = tid & 31;
  const int wave = tid >> 5;
  const int half = lane >> 4;
  const int idx  = lane & 15;

  float hd   = head_decay[h];
  float raw  = 0.3f + 0.65f * sigmoidf_(hd);
  float dec  = fminf(fmaxf(raw, 1e-5f), 0.999f);
  float dtv  = 1.0f - dec;
  float logd = logf(dec);

  for (int j = tid; j < LDS_C_FLOATS; j += 256) Cbuf[j] = 0.0f;
  if (tid < DHH) Zb[tid] = 0.0f;
  __syncthreads();

  for (int c = 0; c < LL / CT; ++c) {
    int c0  = c * CT;
    int bl0 = b * LL + c0;

    // ---- Phase A: stage Q/K/V chunk + per-position coefficients ----
#ifdef USE_TDM
    if (tid < 32) {   // one wave issues the three tensor DMAs
      tdm_load_tile(q_f + ((size_t)bl0 * HH + h) * DHH, LDS_QB_OFF_B, HH * DHH);
      tdm_load_tile(k_f + ((size_t)bl0 * HH + h) * DHH, LDS_KB_OFF_B, HH * DHH);
      tdm_load_tile(qkv + (size_t)bl0 * (3 * DD) + 2 * DD + h * DHH, LDS_VB_OFF_B, 3 * DD);
      __builtin_amdgcn_s_wait_tensorcnt(0);
    }
#else
    for (int j = tid; j < CT * DHH; j += 256) {
      int r = j >> 7, d = j & (DHH - 1);
      Qb[j] = q_f[((size_t)(bl0 + r) * HH + h) * DHH + d];
      Kb[j] = k_f[((size_t)(bl0 + r) * HH + h) * DHH + d];
      Vb[j] = qkv[(size_t)(bl0 + r) * (3 * DD) + 2 * DD + h * DHH + d];
    }
#endif
    if (tid < CT) {
      int tg = c0 + tid;
      float g    = gate[(b * LL + tg) * HH + h];
      float df   = __expf((float)(tg + 1) * logd);
      float safe = df + 1e-8f;
      cKV[tid] = g * dtv / safe;
      cZ[tid]  = dtv / safe;
      dfb[tid] = df;
    }
    __syncthreads();

    // ---- Phase B: A = Q @ K^T (16x16, K=128), wave 0 ----
    if (wave == 0) {
      v8f acc = {};
      for (int k0 = 0; k0 < DHH; k0 += 4) {
        int ka = k0 + half * 2;
        v2f a;  a.x  = Qb[idx * DHH + ka]; a.y  = Qb[idx * DHH + ka + 1];
        v2f bb; bb.x = Kb[idx * DHH + ka]; bb.y = Kb[idx * DHH + ka + 1];
        acc = __builtin_amdgcn_wmma_f32_16x16x4_f32(false, a, false, bb, (short)0, acc, false, false);
      }
#pragma unroll
      for (int r = 0; r < 8; ++r) Ab[(r + half * 8) * CT + idx] = acc[r];
    }
    __syncthreads();

    // ---- Phase C: causal-masked W, denominator ----
    {
      int t = tid >> 4, s = tid & 15;
      Wb[tid] = (s <= t) ? Ab[tid] * cKV[s] : 0.0f;
    }
    if (tid < CT) {
      int t = tid;
      float qz = 0.0f;
      for (int d = 0; d < DHH; ++d) qz += Qb[t * DHH + d] * Zb[d];
      float di = 0.0f;
      for (int s = 0; s <= t; ++s) di += Ab[t * CT + s] * cZ[s];
      denb[t] = fmaxf(dfb[t] * (qz + di), 1e-5f);
    }
    __syncthreads();

    // ---- Phase D: out = df*(Q@C_prev + W@V)/den ; wave -> 16-col tile ----
    {
      int e0 = wave * 16;
      v8f acc = {};
      for (int k0 = 0; k0 < DHH; k0 += 4) {          // inter-chunk: Q(16x128)@C(128x128)
        int ka = k0 + half * 2;
        v2f a;  a.x  = Qb[idx * DHH + ka];           a.y  = Qb[idx * DHH + ka + 1];
        v2f bb; bb.x = Cbuf[ka * DHH + e0 + idx];    bb.y = Cbuf[(ka + 1) * DHH + e0 + idx];
        acc = __builtin_amdgcn_wmma_f32_16x16x4_f32(false, a, false, bb, (short)0, acc, false, false);
      }
#pragma unroll
      for (int k0 = 0; k0 < CT; k0 += 4) {           // intra-chunk: W(16x16)@V(16x128)
        int ka = k0 + half * 2;
        v2f a;  a.x  = Wb[idx * CT + ka];            a.y  = Wb[idx * CT + ka + 1];
        v2f bb; bb.x = Vb[ka * DHH + e0 + idx];      bb.y = Vb[(ka + 1) * DHH + e0 + idx];
        acc = __builtin_amdgcn_wmma_f32_16x16x4_f32(false, a, false, bb, (short)0, acc, false, false);
      }
#pragma unroll
      for (int r = 0; r < 8; ++r) {
        int t = r + half * 8;
        attn_raw[((size_t)(b * LL + c0 + t)) * DD + h * DHH + e0 + idx]
            = dfb[t] * acc[r] / denb[t];
      }
    }
    __syncthreads();

    // ---- Phase E: C += K'^T(128x16) @ V(16x128); Z += coefZ^T K ----
    {
      int d0 = wave * 16;
      for (int et = 0; et < 8; ++et) {
        int e0 = et * 16;
        v8f acc;
#pragma unroll
        for (int r = 0; r < 8; ++r) acc[r] = Cbuf[(d0 + r + half * 8) * DHH + e0 + idx];
#pragma unroll
        for (int k0 = 0; k0 < CT; k0 += 4) {
          int ka = k0 + half * 2;
          v2f a;  a.x  = cKV[ka]     * Kb[ka * DHH + d0 + idx];
                  a.y  = cKV[ka + 1] * Kb[(ka + 1) * DHH + d0 + idx];
          v2f bb; bb.x = Vb[ka * DHH + e0 + idx];   bb.y = Vb[(ka + 1) * DHH + e0 + idx];
          acc = __builtin_amdgcn_wmma_f32_16x16x4_f32(false, a, false, bb, (short)0, acc, false, false);
        }
#pragma unroll
        for (int r = 0; r < 8; ++r) Cbuf[(d0 + r + half * 8) * DHH + e0 + idx] = acc[r];
      }
      if (tid < DHH) {
        float z = Zb[tid];
        for (int s = 0; s < CT; ++s) z += cZ[s] * Kb[s * DHH + tid];
        Zb[tid] = z;
      }
    }
    __syncthreads();
  }
}

// ---------------------------------------------------------------------------
// Host-side launch
// ---------------------------------------------------------------------------
extern "C" void kernel_launch(void* const* d_in, const int* in_sizes, int n_in,
                              void* d_out, int out_size, void* d_ws, size_t ws_size,
                              hipStream_t stream) {
  (void)in_sizes; (void)n_in; (void)out_size; (void)ws_size;

  const float* x          = (const float*)d_in[0];
  const float* ln_w       = (const float*)d_in[1];
  const float* Wc         = (const float*)d_in[2];
  const float* Wqkv       = (const float*)d_in[3];
  const float* Wreso      = (const float*)d_in[4];
  const float* qn_w       = (const float*)d_in[5];
  const float* kn_w       = (const float*)d_in[6];
  const float* Wg         = (const float*)d_in[7];
  const float* head_decay = (const float*)d_in[8];
  const float* temp       = (const float*)d_in[9];
  const float* Wp         = (const float*)d_in[10];
  const float* mem_w      = (const float*)d_in[11];
  float* out = (float*)d_out;

  const int M = BB * LL;  // 1024 rows

  float* ws = (float*)d_ws;
  float* x_norm    = ws;
  float* latent    = x_norm    + (size_t)M * DD;
  float* qkv       = latent    + (size_t)M * LAT;
  float* q_f       = qkv       + (size_t)M * 3 * DD;
  float* k_f       = q_f       + (size_t)M * DD;
  float* params    = k_f       + (size_t)M * DD;
  float* gate      = params    + (size_t)M * 16;
  float* attn_raw  = gate      + (size_t)M * HH;
  float* attn_norm = attn_raw  + (size_t)M * DD;
  float* gate_val  = attn_norm + (size_t)M * DD;

  // 1) x_norm = rmsnorm(x, ln_w)
  rmsnorm_rows_kernel<<<M, 256, 0, stream>>>(x, ln_w, x_norm, DD);

  // 2) latent = silu(x_norm @ Wc)   1024x256, K=512
  gemm_wmma_f32_kernel<<<128, 256, 0, stream>>>(x_norm, Wc, latent, nullptr, M, LAT, DD, 1);

  // 3) qkv = latent @ Wqkv          1024x1536, K=256
  gemm_wmma_f32_kernel<<<768, 256, 0, stream>>>(latent, Wqkv, qkv, nullptr, M, 3 * DD, LAT, 0);

  // 4) params = latent @ Wreso      1024x16, K=256
  gemm_wmma_f32_kernel<<<8, 256, 0, stream>>>(latent, Wreso, params, nullptr, M, 16, LAT, 0);

  // 5) gate_val = silu(latent @ Wg) 1024x512, K=256
  gemm_wmma_f32_kernel<<<256, 256, 0, stream>>>(latent, Wg, gate_val, nullptr, M, DD, LAT, 1);

  // 6) q/k: rmsnorm + rope + elu+1
  qk_rope_kernel<<<dim3(M * HH, 2), 128, 0, stream>>>(qkv, qn_w, kn_w, q_f, k_f);

  // 7) resonance gate
  gate_kernel<<<(M * HH + 255) / 256, 256, 0, stream>>>(params, temp, gate);

  // 8) chunked WMMA recurrence, one workgroup per (b,h)
  size_t smem = (size_t)REC_SMEM_FLOATS * sizeof(float);
  recurrence_chunked_kernel<<<BB * HH, 256, smem, stream>>>(q_f, k_f, qkv, gate,
                                                            head_decay, attn_raw);

  // 9) attn_norm = rmsnorm(attn_raw, mem_w)
  rmsnorm_rows_kernel<<<M, 256, 0, stream>>>(attn_raw, mem_w, attn_norm, DD);

  // 10) out = (attn_norm @ Wp) * gate_val   1024x512, K=512
  gemm_wmma_f32_kernel<<<256, 256, 0, stream>>>(attn_norm, Wp, out, gate_val, M, DD, DD, 2);
}